// CausalAttGCNConv_90555090468952
// MI455X (gfx1250) — compile-verified
//
#include <hip/hip_runtime.h>
#include <math.h>

#define N_NODES 50000
#define N_EDGES 800000
#define IN_CH   128
#define HC      128   // OUT_CH * HEADS
#define OUT_CH  64
#define LDA     132   // padded LDS row stride (528B = 33*16B: b128-aligned, bank-rotating)

typedef __attribute__((ext_vector_type(2))) float v2f;
typedef __attribute__((ext_vector_type(8))) float v8f;
typedef __attribute__((__vector_size__(16))) int v4i_t;

// 16B global -> LDS async copy (ASYNCcnt-tracked); layered fallbacks.
__device__ __forceinline__ void async_copy_b128(const float* gsrc, float* ldst) {
#if __has_builtin(__builtin_amdgcn_global_load_async_to_lds_b128)
  __builtin_amdgcn_global_load_async_to_lds_b128((v4i_t*)gsrc, (v4i_t*)ldst, 0, 0);
#elif defined(__gfx1250__)
  asm volatile("global_load_async_to_lds_b128 %0, %1, off"
               :: "v"((unsigned int)(uintptr_t)ldst), "v"(gsrc) : "memory");
#else
  *(float4*)ldst = *(const float4*)gsrc;
#endif
}

__device__ __forceinline__ void async_wait0() {
#if __has_builtin(__builtin_amdgcn_s_wait_asynccnt)
  __builtin_amdgcn_s_wait_asynccnt(0);
#elif defined(__gfx1250__)
  asm volatile("s_wait_asynccnt 0" ::: "memory");
#endif
}

// ---------------------------------------------------------------------------
// K1: xp = x @ W_lin   [N,128] x [128,128] -> [N,128], f32 WMMA 16x16x4
// One block (8 waves) per 16-row tile; A tile staged once in LDS via async
// load (removes 8x redundant global reads of x); wave w owns col tile w.
// ---------------------------------------------------------------------------
__global__ __launch_bounds__(256) void proj_gemm_kernel(
    const float* __restrict__ x, const float* __restrict__ W,
    float* __restrict__ xp) {
  __shared__ float ldsA[16 * LDA];
  const int tid     = threadIdx.x;
  const int lane    = tid & 31;
  const int colTile = tid >> 5;      // 8 col tiles of 16
  const int rowTile = blockIdx.x;    // 3125 row tiles

  // stage 16x128 A tile (8KB = 512 float4, 2 per thread)
#pragma unroll
  for (int i = 0; i < 2; ++i) {
    const int idx = tid + i * 256;   // float4 index
    const int row = idx >> 5;        // 32 float4 per row
    const int c4  = idx & 31;
    async_copy_b128(x + (size_t)(rowTile * 16 + row) * IN_CH + c4 * 4,
                    &ldsA[row * LDA + c4 * 4]);
  }
  async_wait0();
  __syncthreads();

  const int m     = lane & 15;
  const int khalf = (lane >> 4) << 1;          // 0 or 2
  const float* arow = &ldsA[m * LDA];
  const float* bcol = W + colTile * 16 + (lane & 15);

  v8f c = {};
  for (int k = 0; k < IN_CH; k += 4) {
    __builtin_prefetch(&bcol[(size_t)(k + khalf + 8) * HC], 0, 1);
    v2f a, b;
    a.x = arow[k + khalf];
    a.y = arow[k + khalf + 1];
    b.x = bcol[(size_t)(k + khalf) * HC];
    b.y = bcol[(size_t)(k + khalf + 1) * HC];
    c = __builtin_amdgcn_wmma_f32_16x16x4_f32(false, a, false, b,
                                              (short)0, c, false, false);
  }

  const int mBase = (lane >> 4) * 8;
  const int n     = lane & 15;
#pragma unroll
  for (int r = 0; r < 8; ++r) {
    xp[(size_t)(rowTile * 16 + mBase + r) * HC + colTile * 16 + n] = c[r];
  }
}

// ---------------------------------------------------------------------------
// K2: node_score[n] = 0.5 * dot(xp[n,:], att_flat)   (heads fold flat)
// ---------------------------------------------------------------------------
__global__ __launch_bounds__(256) void node_score_kernel(
    const float* __restrict__ xp, const float* __restrict__ att,
    float* __restrict__ score) {
  int n = blockIdx.x * blockDim.x + threadIdx.x;
  if (n >= N_NODES) return;
  const float4* xr = (const float4*)(xp + (size_t)n * HC);
  const float4* ar = (const float4*)att;
  float s = 0.0f;
#pragma unroll 8
  for (int i = 0; i < HC / 4; ++i) {
    float4 xv = xr[i], av = ar[i];
    s += xv.x * av.x + xv.y * av.y + xv.z * av.z + xv.w * av.w;
  }
  score[n] = 0.5f * s;
}

// ---------------------------------------------------------------------------
// K3/K5: partial max / partial exp-sum over edge scores (256 blocks)
// ---------------------------------------------------------------------------
__global__ __launch_bounds__(256) void edge_max_kernel(
    const int* __restrict__ ei, const float* __restrict__ score,
    float* __restrict__ part) {
  __shared__ float sm[256];
  float m = -3.0e38f;
  for (int e = blockIdx.x * blockDim.x + threadIdx.x; e < N_EDGES;
       e += gridDim.x * blockDim.x) {
    float s = score[ei[e]] + score[ei[N_EDGES + e]];
    m = fmaxf(m, s);
  }
  sm[threadIdx.x] = m;
  __syncthreads();
  for (int off = 128; off > 0; off >>= 1) {
    if ((int)threadIdx.x < off)
      sm[threadIdx.x] = fmaxf(sm[threadIdx.x], sm[threadIdx.x + off]);
    __syncthreads();
  }
  if (threadIdx.x == 0) part[blockIdx.x] = sm[0];
}

__global__ __launch_bounds__(256) void final_max_kernel(
    const float* __restrict__ part, float* __restrict__ gv) {
  __shared__ float sm[256];
  sm[threadIdx.x] = part[threadIdx.x];
  __syncthreads();
  for (int off = 128; off > 0; off >>= 1) {
    if ((int)threadIdx.x < off)
      sm[threadIdx.x] = fmaxf(sm[threadIdx.x], sm[threadIdx.x + off]);
    __syncthreads();
  }
  if (threadIdx.x == 0) gv[0] = sm[0];
}

__global__ __launch_bounds__(256) void edge_sum_kernel(
    const int* __restrict__ ei, const float* __restrict__ score,
    const float* __restrict__ gv, float* __restrict__ part) {
  __shared__ float sm[256];
  const float gmax = gv[0];
  float acc = 0.0f;
  for (int e = blockIdx.x * blockDim.x + threadIdx.x; e < N_EDGES;
       e += gridDim.x * blockDim.x) {
    float s = score[ei[e]] + score[ei[N_EDGES + e]];
    acc += expf(s - gmax);
  }
  sm[threadIdx.x] = acc;
  __syncthreads();
  for (int off = 128; off > 0; off >>= 1) {
    if ((int)threadIdx.x < off) sm[threadIdx.x] += sm[threadIdx.x + off];
    __syncthreads();
  }
  if (threadIdx.x == 0) part[blockIdx.x] = sm[0];
}

__global__ __launch_bounds__(256) void final_sum_kernel(
    const float* __restrict__ part, float* __restrict__ gv) {
  __shared__ float sm[256];
  sm[threadIdx.x] = part[threadIdx.x];
  __syncthreads();
  for (int off = 128; off > 0; off >>= 1) {
    if ((int)threadIdx.x < off) sm[threadIdx.x] += sm[threadIdx.x + off];
    __syncthreads();
  }
  if (threadIdx.x == 0) gv[1] = sm[0];
}

// ---------------------------------------------------------------------------
// K0: zero the aggregation buffer
// ---------------------------------------------------------------------------
__global__ __launch_bounds__(256) void zero_kernel(float* __restrict__ p, int n) {
  for (int i = blockIdx.x * blockDim.x + threadIdx.x; i < n;
       i += gridDim.x * blockDim.x)
    p[i] = 0.0f;
}

// ---------------------------------------------------------------------------
// K7: agg[row] += w_e * xp[col]   one wave per edge; lane owns 4 channels
// (128 ch = 32 lanes x float4). xp is L2-resident (25.6MB << 192MB).
// ---------------------------------------------------------------------------
__global__ __launch_bounds__(256) void scatter_kernel(
    const int* __restrict__ ei, const float* __restrict__ score,
    const float* __restrict__ xp, const float* __restrict__ gv,
    float* __restrict__ agg) {
  const int lane = threadIdx.x & 31;
  const int e    = blockIdx.x * (blockDim.x >> 5) + (threadIdx.x >> 5);
  if (e >= N_EDGES) return;
  const int row = ei[e];
  const int col = ei[N_EDGES + e];
  const float w = expf(score[row] + score[col] - gv[0]) / gv[1];
  const float4 v = ((const float4*)(xp + (size_t)col * HC))[lane];
  float* dst = agg + (size_t)row * HC + lane * 4;
  atomicAdd(dst + 0, w * v.x);
  atomicAdd(dst + 1, w * v.y);
  atomicAdd(dst + 2, w * v.z);
  atomicAdd(dst + 3, w * v.w);
}

// ---------------------------------------------------------------------------
// K8: out = tanh(agg @ W_out + b)   [N,128] x [128,64], f32 WMMA 16x16x4
// One block (4 waves) per 16-row tile; A tile staged in LDS (async).
// ---------------------------------------------------------------------------
__global__ __launch_bounds__(128) void out_gemm_kernel(
    const float* __restrict__ agg, const float* __restrict__ W,
    const float* __restrict__ bias, float* __restrict__ out) {
  __shared__ float ldsA[16 * LDA];
  const int tid     = threadIdx.x;
  const int lane    = tid & 31;
  const int colTile = tid >> 5;      // 4 col tiles of 16
  const int rowTile = blockIdx.x;    // 3125 row tiles

#pragma unroll
  for (int i = 0; i < 4; ++i) {
    const int idx = tid + i * 128;   // 512 float4 total
    const int row = idx >> 5;
    const int c4  = idx & 31;
    async_copy_b128(agg + (size_t)(rowTile * 16 + row) * HC + c4 * 4,
                    &ldsA[row * LDA + c4 * 4]);
  }
  async_wait0();
  __syncthreads();

  const int m     = lane & 15;
  const int khalf = (lane >> 4) << 1;
  const float* arow = &ldsA[m * LDA];
  const float* bcol = W + colTile * 16 + (lane & 15);

  v8f c = {};
  for (int k = 0; k < HC; k += 4) {
    v2f a, b;
    a.x = arow[k + khalf];
    a.y = arow[k + khalf + 1];
    b.x = bcol[(size_t)(k + khalf) * OUT_CH];
    b.y = bcol[(size_t)(k + khalf + 1) * OUT_CH];
    c = __builtin_amdgcn_wmma_f32_16x16x4_f32(false, a, false, b,
                                              (short)0, c, false, false);
  }

  const int mBase = (lane >> 4) * 8;
  const int n     = lane & 15;
  const float bv  = bias[colTile * 16 + n];
#pragma unroll
  for (int r = 0; r < 8; ++r) {
    out[(size_t)(rowTile * 16 + mBase + r) * OUT_CH + colTile * 16 + n] =
        tanhf(c[r] + bv);
  }
}

// ---------------------------------------------------------------------------
extern "C" void kernel_launch(void* const* d_in, const int* in_sizes, int n_in,
                              void* d_out, int out_size, void* d_ws,
                              size_t ws_size, hipStream_t stream) {
  const float* x     = (const float*)d_in[0];
  const int*   ei    = (const int*)d_in[1];   // [2, E]
  const float* W_lin = (const float*)d_in[2]; // [128,128]
  const float* att   = (const float*)d_in[3]; // [128] flat
  const float* W_out = (const float*)d_in[4]; // [128,64]
  const float* b_out = (const float*)d_in[5]; // [64]
  float* out = (float*)d_out;

  // workspace layout
  float* xp    = (float*)d_ws;                       // N*128
  float* agg   = xp + (size_t)N_NODES * HC;          // N*128
  float* score = agg + (size_t)N_NODES * HC;         // N
  float* part  = score + N_NODES;                    // 256
  float* gv    = part + 256;                         // 2 (gmax, gsum)

  // 1) projection GEMM: one block per 16-row tile
  proj_gemm_kernel<<<N_NODES / 16, 256, 0, stream>>>(x, W_lin, xp);
  // 2) per-node attention score
  node_score_kernel<<<(N_NODES + 255) / 256, 256, 0, stream>>>(xp, att, score);
  // 3-6) global softmax normalizers over all edges
  edge_max_kernel<<<256, 256, 0, stream>>>(ei, score, part);
  final_max_kernel<<<1, 256, 0, stream>>>(part, gv);
  edge_sum_kernel<<<256, 256, 0, stream>>>(ei, score, gv, part);
  final_sum_kernel<<<1, 256, 0, stream>>>(part, gv);
  // 7) zero agg then scatter-add weighted messages (1 wave per edge)
  zero_kernel<<<1024, 256, 0, stream>>>(agg, N_NODES * HC);
  scatter_kernel<<<N_EDGES / 8, 256, 0, stream>>>(ei, score, xp, gv, agg);
  // 8) output GEMM + bias + tanh: one block per 16-row tile
  out_gemm_kernel<<<N_NODES / 16, 128, 0, stream>>>(agg, W_out, b_out, out);
}